// SimCLR_58042188038372
// MI455X (gfx1250) — compile-verified
//
#include <hip/hip_runtime.h>
#include <hip/hip_bf16.h>
#include <math.h>

typedef __attribute__((ext_vector_type(2))) float v2f;
typedef __attribute__((ext_vector_type(8))) float v8f;

#define NROWS 8192
#define KDIM  1024
#define JDIM  4000
#define KSEG  4            // K-split for memory-level parallelism in K1
#define KLEN  (KDIM / KSEG)

// workspace layout (float offsets)
#define OFF_WPAD   0        // 1024*16 = 16384 : W1 padded to 16 cols
#define OFF_Y      16384    // KSEG*2*8192*3 = 196608 : partial y per K-segment
#define OFF_H      212992   // 2*8192*3 = 49152 : h = relu(bn1(y))
#define OFF_AB1    262144   // 12  : BN1 affine A,B per branch/col
#define OFF_HSTAT  262176   // 24  : hbar[2][3], cov[2][3][3]
#define OFF_TBL    262208   // 2*4000*8 = 64000 : {w0,w1,w2,alpha | beta,u,p,q}
#define OFF_ROWOUT 326208   // 8*8192 : per-row {nrm,du,dp,dq} x 2 branches
#define OFF_SIMS   391744   // 3*8192 : simcol, simrow0, simrow1  (ends 416320)

// ---------------- K0: pad W1 (1024x3) -> Wpad (1024x16) ----------------
__global__ void k0_pad_w1(const float* __restrict__ W1, float* __restrict__ ws) {
  int idx = blockIdx.x * blockDim.x + threadIdx.x;   // 0..16383
  if (idx < KDIM * 16) {
    int k = idx >> 4, c = idx & 15;
    ws[OFF_WPAD + idx] = (c < 3) ? W1[k * 3 + c] : 0.0f;
  }
}

// ---------------- K1: y = x @ W1 via V_WMMA_F32_16X16X4_F32 ----------------
// 4096 waves: (seg, branch, tile). Each wave: 16-row tile over K-segment of 256.
// A (16x4 f32): lane L (0..15): row=L, holds K=k0+0,k0+1 ; lane L+16: K=k0+2,k0+3.
// B (4x16 f32): mirrored: lanes 0..15 hold rows K=k0+0/k0+1, lanes 16..31 K=k0+2/k0+3.
__global__ void k1_gemm1_wmma(const float* __restrict__ x1,
                              const float* __restrict__ x2,
                              float* __restrict__ ws) {
  const float* Wp = ws + OFF_WPAD;
  float* yp = ws + OFF_Y;
  int lane = threadIdx.x & 31;
  int wave = (blockIdx.x * blockDim.x + threadIdx.x) >> 5;  // 0..4095
  int seg  = wave >> 10;                                    // K segment
  int rem  = wave & 1023;
  int b    = rem >> 9;                                      // branch
  int tile = rem & 511;                                     // 16-row tile
  const float* x = b ? x2 : x1;
  int mcol = lane & 15;                 // A row / B col
  int kb   = (lane >> 4) << 1;          // 0 or 2
  int kbase = seg * KLEN;
  const float* xrow = x + (size_t)(tile * 16 + mcol) * KDIM + kbase + kb;
  const float* wcol = Wp + mcol + (kbase + kb) * 16;

  v8f acc = {0.f, 0.f, 0.f, 0.f, 0.f, 0.f, 0.f, 0.f};
#pragma unroll 8
  for (int k0 = 0; k0 < KLEN; k0 += 4) {
    v2f a  = { xrow[k0],        xrow[k0 + 1] };          // x[row][kbase+k0+kb ..]
    v2f bb = { wcol[k0 * 16],   wcol[(k0 + 1) * 16] };   // Wpad[kbase+k0+kb..][col]
    acc = __builtin_amdgcn_wmma_f32_16x16x4_f32(
        false, a, false, bb, (short)0, acc, false, false);
  }
  // D: VGPR r -> lanes 0..15: M=r, lanes 16..31: M=r+8; N = lane&15
  if (mcol < 3) {
    int mbase = (lane >> 4) << 3;
#pragma unroll
    for (int r = 0; r < 8; ++r) {
      int row = tile * 16 + mbase + r;
      yp[(((size_t)seg * 2 + b) * NROWS + row) * 3 + mcol] = acc[r];
    }
  }
}

// ---------------- K2: combine K-segments; BN1 stats -> affine A,B ----------------
// Writes combined y into segment-0 slot (each (i,b,c) touched by exactly one thread).
__global__ void k2_bn1(const float* __restrict__ g1, const float* __restrict__ be1,
                       float* __restrict__ ws) {
  __shared__ float red[1024];
  float* y = ws + OFF_Y;
  int tid = threadIdx.x;
  float s[2][3] = {}, sq[2][3] = {};
  for (int i = tid; i < NROWS; i += 1024)
    for (int b = 0; b < 2; ++b)
      for (int c = 0; c < 3; ++c) {
        float v = 0.0f;
        for (int seg = 0; seg < KSEG; ++seg)
          v += y[(((size_t)seg * 2 + b) * NROWS + i) * 3 + c];
        y[((size_t)b * NROWS + i) * 3 + c] = v;   // combined into seg-0 slot
        s[b][c] += v; sq[b][c] += v * v;
      }
  float tot[12];
  for (int q = 0; q < 12; ++q) {
    int b = q / 6, t = (q % 6) / 3, c = q % 3;
    red[tid] = t ? sq[b][c] : s[b][c];
    __syncthreads();
    for (int off = 512; off; off >>= 1) { if (tid < off) red[tid] += red[tid + off]; __syncthreads(); }
    tot[q] = red[0];
    __syncthreads();
  }
  if (tid == 0)
    for (int b = 0; b < 2; ++b)
      for (int c = 0; c < 3; ++c) {
        float m   = tot[b * 6 + c] / (float)NROWS;
        float var = tot[b * 6 + 3 + c] / (float)NROWS - m * m;
        float inv = 1.0f / sqrtf(var + 1e-5f);
        float A = g1[c] * inv;
        ws[OFF_AB1 + b * 6 + c]     = A;
        ws[OFF_AB1 + b * 6 + 3 + c] = be1[c] - A * m;   // b1 cancels inside BN
      }
}

// ---------------- K3: h = relu(bn1(y)); mean + second-moment of h ----------------
__global__ void k3_h(float* __restrict__ ws) {
  __shared__ float red[1024];
  const float* y  = ws + OFF_Y;      // combined y lives in seg-0 slot
  const float* AB = ws + OFF_AB1;
  float* h = ws + OFF_H;
  int tid = threadIdx.x;
  float A[2][3], B[2][3];
  for (int b = 0; b < 2; ++b)
    for (int c = 0; c < 3; ++c) { A[b][c] = AB[b * 6 + c]; B[b][c] = AB[b * 6 + 3 + c]; }
  float acc[18] = {};
  for (int i = tid; i < NROWS; i += 1024)
    for (int b = 0; b < 2; ++b) {
      float hv[3];
      for (int c = 0; c < 3; ++c) {
        float v = fmaf(A[b][c], y[((size_t)b * NROWS + i) * 3 + c], B[b][c]);
        hv[c] = fmaxf(v, 0.0f);
        h[((size_t)b * NROWS + i) * 3 + c] = hv[c];
      }
      acc[b*9+0]+=hv[0]; acc[b*9+1]+=hv[1]; acc[b*9+2]+=hv[2];
      acc[b*9+3]+=hv[0]*hv[0]; acc[b*9+4]+=hv[1]*hv[1]; acc[b*9+5]+=hv[2]*hv[2];
      acc[b*9+6]+=hv[0]*hv[1]; acc[b*9+7]+=hv[0]*hv[2]; acc[b*9+8]+=hv[1]*hv[2];
    }
  float tot[18];
  for (int q = 0; q < 18; ++q) {
    red[tid] = acc[q]; __syncthreads();
    for (int off = 512; off; off >>= 1) { if (tid < off) red[tid] += red[tid + off]; __syncthreads(); }
    tot[q] = red[0]; __syncthreads();
  }
  if (tid == 0) {
    const float invN = 1.0f / (float)NROWS;
    for (int b = 0; b < 2; ++b) {
      float hb[3] = { tot[b*9+0]*invN, tot[b*9+1]*invN, tot[b*9+2]*invN };
      for (int c = 0; c < 3; ++c) ws[OFF_HSTAT + b * 3 + c] = hb[c];
      float* cv = ws + OFF_HSTAT + 6 + b * 9;
      float C00 = tot[b*9+3]*invN - hb[0]*hb[0];
      float C11 = tot[b*9+4]*invN - hb[1]*hb[1];
      float C22 = tot[b*9+5]*invN - hb[2]*hb[2];
      float C01 = tot[b*9+6]*invN - hb[0]*hb[1];
      float C02 = tot[b*9+7]*invN - hb[0]*hb[2];
      float C12 = tot[b*9+8]*invN - hb[1]*hb[2];
      cv[0]=C00; cv[1]=C01; cv[2]=C02; cv[3]=C01; cv[4]=C11; cv[5]=C12; cv[6]=C02; cv[7]=C12; cv[8]=C22;
    }
  }
}

// ---------------- K4: per-column BN2 affine + special rows u/p/q ----------------
// var(h@w_j) = w^T Cov(h) w ; mean part via hbar. b2 cancels inside BN.
__global__ void k4_tbl(const float* __restrict__ W2, const float* __restrict__ g2,
                       const float* __restrict__ be2, float* __restrict__ ws) {
  int idx = blockIdx.x * blockDim.x + threadIdx.x;
  if (idx >= 2 * JDIM) return;
  int b = idx / JDIM, j = idx % JDIM;
  float w0 = W2[j], w1 = W2[JDIM + j], w2 = W2[2 * JDIM + j];
  const float* hb = ws + OFF_HSTAT + b * 3;
  const float* C  = ws + OFF_HSTAT + 6 + b * 9;
  float mu  = hb[0]*w0 + hb[1]*w1 + hb[2]*w2;
  float var = C[0]*w0*w0 + C[4]*w1*w1 + C[8]*w2*w2
            + 2.0f*(C[1]*w0*w1 + C[2]*w0*w2 + C[5]*w1*w2);
  float alpha = g2[j] * (1.0f / sqrtf(var + 1e-5f));
  float beta  = be2[j] - alpha * mu;
  const float* h = ws + OFF_H + (size_t)b * NROWS * 3;
  float sL = h[8191*3+0]*w0 + h[8191*3+1]*w1 + h[8191*3+2]*w2;
  float s0 = h[0]*w0 + h[1]*w1 + h[2]*w2;
  float s1 = h[3]*w0 + h[4]*w1 + h[5]*w2;
  float* t = ws + OFF_TBL + (size_t)(b * JDIM + j) * 8;
  t[0]=w0; t[1]=w1; t[2]=w2; t[3]=alpha;
  t[4]=beta;
  t[5]=fmaxf(fmaf(alpha,sL,beta),0.f);   // u = z[N-1]
  t[6]=fmaxf(fmaf(alpha,s0,beta),0.f);   // p = z[0]
  t[7]=fmaxf(fmaf(alpha,s1,beta),0.f);   // q = z[1]
}

// ---------------- K5: per row: ||z||^2, z.u, z.p, z.q ----------------
#define RPT 16
__global__ void k5_main(float* __restrict__ ws) {
  int lane = threadIdx.x & 31;
  int wave = (blockIdx.x * blockDim.x + threadIdx.x) >> 5;   // 0..1023
  int b = wave >> 9;
  int rowBase = (wave & 511) * RPT;
  const float*  h = ws + OFF_H + (size_t)b * NROWS * 3;
  const float4* T = reinterpret_cast<const float4*>(ws + OFF_TBL) + (size_t)b * JDIM * 2;
  float h0[RPT], h1[RPT], h2[RPT];
#pragma unroll
  for (int r = 0; r < RPT; ++r) {
    h0[r]=h[(rowBase+r)*3+0]; h1[r]=h[(rowBase+r)*3+1]; h2[r]=h[(rowBase+r)*3+2];
  }
  float nrm[RPT]={}, du[RPT]={}, dp[RPT]={}, dq[RPT]={};
  for (int j = lane; j < JDIM; j += 32) {
    float4 t0 = T[j*2], t1 = T[j*2+1];
#pragma unroll
    for (int r = 0; r < RPT; ++r) {
      float s = fmaf(h0[r], t0.x, fmaf(h1[r], t0.y, h2[r]*t0.z));
      float z = fmaxf(fmaf(t0.w, s, t1.x), 0.0f);
      nrm[r]=fmaf(z,z,nrm[r]); du[r]=fmaf(z,t1.y,du[r]);
      dp[r] =fmaf(z,t1.z,dp[r]); dq[r]=fmaf(z,t1.w,dq[r]);
    }
  }
#pragma unroll
  for (int r = 0; r < RPT; ++r)
    for (int m = 16; m; m >>= 1) {
      nrm[r]+=__shfl_xor(nrm[r],m); du[r]+=__shfl_xor(du[r],m);
      dp[r] +=__shfl_xor(dp[r],m);  dq[r]+=__shfl_xor(dq[r],m);
    }
  if (lane == 0) {
    float* ro = ws + OFF_ROWOUT + (size_t)b * 4 * NROWS;
#pragma unroll
    for (int r = 0; r < RPT; ++r) {
      int i = rowBase + r;
      ro[0*NROWS+i]=nrm[r]; ro[1*NROWS+i]=du[r]; ro[2*NROWS+i]=dp[r]; ro[3*NROWS+i]=dq[r];
    }
  }
}

// ---------------- K6a: sim column N-1 and rows 0,1 ----------------
__device__ __forceinline__ void rowstats(const float* n1a, const float* n2a, int i,
                                         float& c1, float& c2, float& rn) {
  float n1 = sqrtf(n1a[i]), n2 = sqrtf(n2a[i]);
  c1 = fmaxf(n1, 1e-12f); c2 = fmaxf(n2, 1e-12f);
  float e1 = n1 / c1, e2 = n2 / c2;
  rn = fmaxf(sqrtf(e1*e1 + e2*e2), 1e-8f);
}

__global__ void k6a_sims(float* __restrict__ ws) {
  const float* ro   = ws + OFF_ROWOUT;
  const float* nrm1 = ro + 0*NROWS; const float* du1 = ro + 1*NROWS;
  const float* dp1  = ro + 2*NROWS; const float* dq1 = ro + 3*NROWS;
  const float* nrm2 = ro + 4*NROWS; const float* du2 = ro + 5*NROWS;
  const float* dp2  = ro + 6*NROWS; const float* dq2 = ro + 7*NROWS;
  float* simcol = ws + OFF_SIMS;
  float* simr0  = simcol + NROWS;
  float* simr1  = simr0 + NROWS;
  float c1L,c2L,rnL, c10,c20,rn0, c11,c21,rn1;
  rowstats(nrm1, nrm2, NROWS-1, c1L, c2L, rnL);
  rowstats(nrm1, nrm2, 0,       c10, c20, rn0);
  rowstats(nrm1, nrm2, 1,       c11, c21, rn1);
  int i = blockIdx.x * blockDim.x + threadIdx.x;
  if (i < NROWS) {
    float c1, c2, rn; rowstats(nrm1, nrm2, i, c1, c2, rn);
    simcol[i] = (du1[i]/(c1*c1L) + du2[i]/(c2*c2L)) / (rn*rnL);
    simr0[i]  = (dp1[i]/(c1*c10) + dp2[i]/(c2*c20)) / (rn*rn0);
    simr1[i]  = (dq1[i]/(c1*c11) + dq2[i]/(c2*c21)) / (rn*rn1);
  }
}

// ---------------- K6b: loss = mean log1p(exp((neg-pos)/T)) ----------------
__global__ void k6b_loss(const float* __restrict__ ws, float* __restrict__ out) {
  __shared__ float red[1024];
  const float* simcol = ws + OFF_SIMS;
  const float* simr0  = simcol + NROWS;
  const float* simr1  = simr0 + NROWS;
  int tid = threadIdx.x;
  float acc = 0.0f;
  for (int k = tid; k < 2*NROWS; k += 1024) {
    float pos = simcol[k & (NROWS-1)];
    float neg;
    if (k < NROWS-1)            neg = simr0[k+1];
    else if (k < 2*NROWS-2)     { int m = k - (NROWS-1); neg = simr1[m==0 ? 0 : m+1]; }
    else if (k == 2*NROWS-2)    neg = simr0[2];      // sim[2,0] == sim[0,2]
    else                        neg = simr1[2];      // sim[2,1] == sim[1,2]
    acc += log1pf(expf((neg - pos) * 0.5f));         // TEMPERATURE = 2
  }
  red[tid] = acc; __syncthreads();
  for (int off = 512; off; off >>= 1) { if (tid < off) red[tid] += red[tid+off]; __syncthreads(); }
  if (tid == 0) out[0] = red[0] / (float)(2*NROWS);
}

extern "C" void kernel_launch(void* const* d_in, const int* in_sizes, int n_in,
                              void* d_out, int out_size, void* d_ws, size_t ws_size,
                              hipStream_t stream) {
  (void)in_sizes; (void)n_in; (void)out_size; (void)ws_size;
  const float* x1  = (const float*)d_in[0];
  const float* x2  = (const float*)d_in[1];
  const float* W1  = (const float*)d_in[2];
  // d_in[3] = b1 : cancels inside BN
  const float* g1  = (const float*)d_in[4];
  const float* be1 = (const float*)d_in[5];
  const float* W2  = (const float*)d_in[6];
  // d_in[7] = b2 : cancels inside BN
  const float* g2  = (const float*)d_in[8];
  const float* be2 = (const float*)d_in[9];
  float* ws  = (float*)d_ws;
  float* out = (float*)d_out;

  k0_pad_w1    <<<64, 256, 0, stream>>>(W1, ws);
  k1_gemm1_wmma<<<512, 256, 0, stream>>>(x1, x2, ws);   // 4096 waves
  k2_bn1       <<<1, 1024, 0, stream>>>(g1, be1, ws);
  k3_h         <<<1, 1024, 0, stream>>>(ws);
  k4_tbl       <<<(2*JDIM + 255)/256, 256, 0, stream>>>(W2, g2, be2, ws);
  k5_main      <<<128, 256, 0, stream>>>(ws);
  k6a_sims     <<<(NROWS + 255)/256, 256, 0, stream>>>(ws);
  k6b_loss     <<<1, 1024, 0, stream>>>(ws, out);
}